// SIAU_26431228739890
// MI455X (gfx1250) — compile-verified
//
#include <hip/hip_runtime.h>
#include <math.h>

// Problem constants (BS, N, C) = (16, 128, 512)
#define BSZ 16
#define NN  128
#define CC  512
#define EPSV 1e-5f

typedef __attribute__((ext_vector_type(2))) float v2f;
typedef __attribute__((ext_vector_type(8))) float v8f;

// Full-precision fp32 WMMA: D(16x16) = A(16x4) * B(4x16) + C
static __device__ __forceinline__ v8f wmma_f32_k4(v2f a, v2f b, v8f c) {
    return __builtin_amdgcn_wmma_f32_16x16x4_f32(
        /*neg_a=*/false, a, /*neg_b=*/false, b,
        /*c_mod=*/(short)0, c, /*reuse_a=*/false, /*reuse_b=*/false);
}

// ---------------------------------------------------------------------------
// Kernel: fold conv_f bias + BN_f into per-output-channel affine (s, t).
// out = relu(gemm * s[co] + t[co])
// ---------------------------------------------------------------------------
__global__ void k_prep(const float* __restrict__ g, const float* __restrict__ bb,
                       const float* __restrict__ m, const float* __restrict__ v,
                       const float* __restrict__ convb,
                       float* __restrict__ sf, float* __restrict__ tf) {
    int co = threadIdx.x + blockIdx.x * blockDim.x;
    if (co < CC) {
        float s = g[co] * rsqrtf(v[co] + EPSV);
        sf[co] = s;
        tf[co] = (convb[co] - m[co]) * s + bb[co];
    }
}

// ---------------------------------------------------------------------------
// Kernel: transpose conv_f_w (CC x 2CC, row-major) -> Wt (2CC x CC) so that
// WMMA B-fragment loads are contiguous across lanes.
// ---------------------------------------------------------------------------
__global__ void k_transpose(const float* __restrict__ Wf, float* __restrict__ Wt) {
    int k = blockIdx.x;            // 0..2*CC-1
    int t = threadIdx.x;           // 0..255
    Wt[(size_t)k * CC + t]        = Wf[(size_t)t * (2 * CC) + k];
    Wt[(size_t)k * CC + t + 256]  = Wf[(size_t)(t + 256) * (2 * CC) + k];
}

// ---------------------------------------------------------------------------
// Kernel: score + softmax.  One block per (b, i) row.
// score[j] = scale * sum_c |x[b,i,c]-x[b,j,c]| * w_diff[c]   (j != i)
// score[i] = -1e8.
// All j-independent terms (y.w_y, conv_e_b, bn_e_m, bn_e_b) drop out of the
// softmax and are omitted.  Writes softmax row W[b,i,:] (128 floats).
// ---------------------------------------------------------------------------
__global__ void k_score_softmax(const float* __restrict__ x,
                                const float* __restrict__ conv_e_w,
                                const float* __restrict__ bn_e_g,
                                const float* __restrict__ bn_e_v,
                                float* __restrict__ Wsm) {
    __shared__ float xi[CC];
    __shared__ float wd[CC];
    __shared__ float sc[NN];
    __shared__ float red[64];

    int bid = blockIdx.x;          // 0..BSZ*NN-1
    int b = bid / NN, i = bid % NN;
    int t = threadIdx.x;           // 256 threads = 8 waves (wave32)

    const float* xrow = x + (size_t)(b * NN + i) * CC;
    for (int q = 0; q < CC / 256; ++q) {
        int idx = t + 256 * q;
        xi[idx] = xrow[idx];
        wd[idx] = conv_e_w[idx];   // first C entries of conv_e_w row 0
    }
    __syncthreads();

    float s_e = bn_e_g[0] * rsqrtf(bn_e_v[0] + EPSV);
    int lane = t & 31, wave = t >> 5;

    // Each wave covers 16 consecutive j rows; lanes parallel over channels.
    for (int mrow = 0; mrow < 16; ++mrow) {
        int j = wave * 16 + mrow;
        const float* xj = x + (size_t)(b * NN + j) * CC;
        float p = 0.f;
        #pragma unroll
        for (int q = 0; q < CC / 32; ++q) {
            int c = lane + 32 * q;             // coalesced 128B per wave
            p += fabsf(xj[c] - xi[c]) * wd[c];
        }
        p += __shfl_xor(p, 16, 32);
        p += __shfl_xor(p,  8, 32);
        p += __shfl_xor(p,  4, 32);
        p += __shfl_xor(p,  2, 32);
        p += __shfl_xor(p,  1, 32);
        if (lane == 0) sc[j] = (j == i) ? -1.0e8f : p * s_e;
    }
    __syncthreads();

    // Block max over 128 values
    if (t < 64) red[t] = fmaxf(sc[t], sc[t + 64]);
    __syncthreads();
    if (t < 32) red[t] = fmaxf(red[t], red[t + 32]);
    __syncthreads();
    if (t < 16) red[t] = fmaxf(red[t], red[t + 16]);
    __syncthreads();
    if (t < 8)  red[t] = fmaxf(red[t], red[t + 8]);
    __syncthreads();
    if (t < 4)  red[t] = fmaxf(red[t], red[t + 4]);
    __syncthreads();
    if (t < 2)  red[t] = fmaxf(red[t], red[t + 2]);
    __syncthreads();
    if (t == 0) red[0] = fmaxf(red[0], red[1]);
    __syncthreads();
    float mx = red[0];
    __syncthreads();

    if (t < NN) sc[t] = __expf(sc[t] - mx);
    __syncthreads();

    // Block sum
    if (t < 64) red[t] = sc[t] + sc[t + 64];
    __syncthreads();
    if (t < 32) red[t] = red[t] + red[t + 32];
    __syncthreads();
    if (t < 16) red[t] = red[t] + red[t + 16];
    __syncthreads();
    if (t < 8)  red[t] = red[t] + red[t + 8];
    __syncthreads();
    if (t < 4)  red[t] = red[t] + red[t + 4];
    __syncthreads();
    if (t < 2)  red[t] = red[t] + red[t + 2];
    __syncthreads();
    if (t == 0) red[0] = red[0] + red[1];
    __syncthreads();
    float inv = 1.0f / red[0];

    if (t < NN) Wsm[(size_t)(b * NN + i) * NN + t] = sc[t] * inv;
}

// ---------------------------------------------------------------------------
// Kernel: x_nb[b] = W[b](128x128) @ x[b](128x512) via V_WMMA_F32_16X16X4_F32.
// Block = 256 thr (8 waves). blockIdx encodes (b, Mtile 0..7, Ngroup 0..3);
// each wave owns one 16x16 output tile.  A strip (16x128) staged in LDS with
// stride-129 padding -> conflict-free fragment reads (bank = (row+k)%64).
// ---------------------------------------------------------------------------
__global__ void k_gemm_xnb(const float* __restrict__ Wsm,
                           const float* __restrict__ x,
                           float* __restrict__ xnb) {
    __shared__ float As[16][129];
    int bid = blockIdx.x;
    int b = bid >> 5;
    int rem = bid & 31;
    int mt = rem >> 2, ng = rem & 3;
    int t = threadIdx.x;

    const float* Wb = Wsm + (size_t)b * NN * NN + (size_t)mt * 16 * NN;
    #pragma unroll
    for (int q = 0; q < 8; ++q) {
        int idx = t + 256 * q;              // 0..2047, coalesced
        int r = idx >> 7, c = idx & 127;
        As[r][c] = Wb[r * NN + c];
    }
    __syncthreads();

    int lane = t & 31, wave = t >> 5;
    int half = lane >> 4, l16 = lane & 15;
    int n0 = (ng * 8 + wave) * 16;
    const float* xb = x + (size_t)b * NN * CC;

    v8f acc = {};
    #pragma unroll 4
    for (int k = 0; k < NN; k += 4) {
        int ka = k + 2 * half;              // lanes 16-31 hold K=2,3 (ISA 7.12.2)
        v2f a, bf;
        a.x  = As[l16][ka];
        a.y  = As[l16][ka + 1];
        bf.x = xb[(size_t)ka * CC + n0 + l16];        // B row K=ka
        bf.y = xb[(size_t)(ka + 1) * CC + n0 + l16];  // B row K=ka+1
        acc = wmma_f32_k4(a, bf, acc);
    }

    int m0 = mt * 16;
    float* ob = xnb + (size_t)b * NN * CC;
    #pragma unroll
    for (int r = 0; r < 8; ++r)             // VGPR r -> rows M=r / M=r+8
        ob[(size_t)(m0 + r + 8 * half) * CC + n0 + l16] = acc[r];
}

// ---------------------------------------------------------------------------
// Kernel: out = relu( (x @ WfA^T + x_nb @ WfB^T) * s + t ).
// K = 1024 split: k<512 reads x, k>=512 reads x_nb.  A chunks (16x256) staged
// in LDS (stride 257 padding); B read coalesced from the pre-transposed Wt.
// ---------------------------------------------------------------------------
__global__ void k_gemm_out(const float* __restrict__ x,
                           const float* __restrict__ xnb,
                           const float* __restrict__ Wt,   // [2*CC][CC]
                           const float* __restrict__ sf,
                           const float* __restrict__ tf,
                           float* __restrict__ out) {
    __shared__ float As[16][257];
    int bid = blockIdx.x;
    int b = bid >> 5;
    int rem = bid & 31;
    int mt = rem >> 2, ng = rem & 3;
    int t = threadIdx.x;
    int lane = t & 31, wave = t >> 5;
    int half = lane >> 4, l16 = lane & 15;
    int n0 = (ng * 8 + wave) * 16;
    int m0 = mt * 16;

    v8f acc = {};
    for (int kc = 0; kc < 2 * CC; kc += 256) {
        const float* src = (kc < CC)
            ? (x   + (size_t)b * NN * CC + kc)
            : (xnb + (size_t)b * NN * CC + (kc - CC));
        __syncthreads();
        #pragma unroll
        for (int q = 0; q < 16; ++q) {
            int idx = t + 256 * q;          // 0..4095, coalesced
            int r = idx >> 8, c = idx & 255;
            As[r][c] = src[(size_t)(m0 + r) * CC + c];
        }
        __syncthreads();
        #pragma unroll 4
        for (int k = 0; k < 256; k += 4) {
            int ka = k + 2 * half;
            v2f a, bf;
            a.x  = As[l16][ka];
            a.y  = As[l16][ka + 1];
            bf.x = Wt[(size_t)(kc + ka) * CC + n0 + l16];
            bf.y = Wt[(size_t)(kc + ka + 1) * CC + n0 + l16];
            acc = wmma_f32_k4(a, bf, acc);
        }
    }

    int co = n0 + l16;
    float s = sf[co], tt = tf[co];
    float* ob = out + (size_t)b * NN * CC;
    #pragma unroll
    for (int r = 0; r < 8; ++r) {
        float v = acc[r] * s + tt;
        ob[(size_t)(m0 + r + 8 * half) * CC + co] = fmaxf(v, 0.0f);
    }
}

// ---------------------------------------------------------------------------
extern "C" void kernel_launch(void* const* d_in, const int* in_sizes, int n_in,
                              void* d_out, int out_size, void* d_ws, size_t ws_size,
                              hipStream_t stream) {
    const float* x        = (const float*)d_in[0];
    // d_in[1] (y), d_in[3] (conv_e_b), d_in[5] (bn_e_b), d_in[6] (bn_e_m):
    // j-independent terms inside the softmax -> provably no effect on output.
    const float* conv_e_w = (const float*)d_in[2];
    const float* bn_e_g   = (const float*)d_in[4];
    const float* bn_e_v   = (const float*)d_in[7];
    const float* conv_f_w = (const float*)d_in[8];
    const float* conv_f_b = (const float*)d_in[9];
    const float* bn_f_g   = (const float*)d_in[10];
    const float* bn_f_b   = (const float*)d_in[11];
    const float* bn_f_m   = (const float*)d_in[12];
    const float* bn_f_v   = (const float*)d_in[13];
    float* out = (float*)d_out;

    // Workspace layout (~7.35 MB of f32):
    float* Wsm = (float*)d_ws;                         // BSZ*NN*NN   (1 MB)
    float* xnb = Wsm + (size_t)BSZ * NN * NN;          // BSZ*NN*CC   (4 MB)
    float* Wt  = xnb + (size_t)BSZ * NN * CC;          // 2*CC*CC     (2 MB)
    float* sf  = Wt  + (size_t)2 * CC * CC;            // CC
    float* tf  = sf  + CC;                             // CC
    (void)ws_size; (void)in_sizes; (void)n_in; (void)out_size;

    k_prep<<<2, 256, 0, stream>>>(bn_f_g, bn_f_b, bn_f_m, bn_f_v, conv_f_b, sf, tf);
    k_transpose<<<2 * CC, 256, 0, stream>>>(conv_f_w, Wt);
    k_score_softmax<<<BSZ * NN, 256, 0, stream>>>(x, conv_e_w, bn_e_g, bn_e_v, Wsm);
    k_gemm_xnb<<<BSZ * 32, 256, 0, stream>>>(Wsm, x, xnb);
    k_gemm_out<<<BSZ * 32, 256, 0, stream>>>(x, xnb, Wt, sf, tf, out);
}